// VladCore_82549271429468
// MI455X (gfx1250) — compile-verified
//
#include <hip/hip_runtime.h>

typedef float v2f __attribute__((ext_vector_type(2)));
typedef float v8f __attribute__((ext_vector_type(8)));
typedef int   v4i __attribute__((vector_size(16)));

#define KROWS 64
#define DCOLS 512
#define TILE_D 128
#define NB 128          // n-block staged in LDS per iteration
#define LNB 132         // padded LDS row stride (132 % 64 == 4 -> conflict-free frag loads)

#if defined(__has_builtin)
# if __has_builtin(__builtin_amdgcn_global_load_async_to_lds_b128)
#  define HAVE_ASYNC_COPY 1
# endif
#endif
#ifndef HAVE_ASYNC_COPY
# define HAVE_ASYNC_COPY 0
#endif

#if HAVE_ASYNC_COPY
typedef __attribute__((address_space(1))) v4i* gv4p;   // global int32x4*
typedef __attribute__((address_space(3))) v4i* lv4p;   // LDS int32x4*
#endif

// ---------------------------------------------------------------------------
// Pass 1: partial[chunk][k][d] = sum_{n in chunk} a_bar[k,n] * x[d,n]
//         asum_p[chunk][k]    = sum_{n in chunk} a_bar[k,n]
// Grid: (D/TILE_D = 4, NS). Block: 256 threads = 8 waves.
// Wave w: ktile = w>>1 (16 k-rows), dtiles = (w&1)*4 .. +3 (4 x 16 d-cols).
// ---------------------------------------------------------------------------
__global__ __launch_bounds__(256) void vlad_gemm_wmma(
    const float* __restrict__ x, const float* __restrict__ a_bar,
    float* __restrict__ partial, float* __restrict__ asum_p,
    int N, int blocksPerChunk)
{
    __shared__ float lA[KROWS * LNB];     // 33792 B
    __shared__ float lX[TILE_D * LNB];    // 67584 B
    __shared__ float lSum[256];

    const int tid   = threadIdx.x;
    const int dtile = blockIdx.x;                 // 0..3
    const int chunk = blockIdx.y;                 // 0..NS-1
    const int d0    = dtile * TILE_D;
    const long nBase = (long)chunk * blocksPerChunk * NB;

    const int lane  = tid & 31;
    const int wave  = tid >> 5;
    const int lhi   = lane >> 4;                  // 0/1 (lane half)
    const int llo   = lane & 15;
    const int kt    = wave >> 1;                  // k-tile 0..3
    const int dquad = (wave & 1) * 4;             // first of 4 d-tiles

    v8f acc0 = {}, acc1 = {}, acc2 = {}, acc3 = {};
    float rowsum = 0.0f;

    for (int blk = 0; blk < blocksPerChunk; ++blk) {
        const long n0 = nBase + (long)blk * NB;

#if HAVE_ASYNC_COPY
        // ---- async DMA fill: global -> LDS directly (ASYNCcnt-tracked) ----
        for (int i = tid; i < KROWS * (NB / 4); i += 256) {
            int r = i >> 5, c4 = (i & 31) << 2;
            const float* g = a_bar + (long)r * N + n0 + c4;
            __builtin_amdgcn_global_load_async_to_lds_b128(
                (gv4p)(void*)g, (lv4p)&lA[r * LNB + c4], 0, 0);
        }
        for (int i = tid; i < TILE_D * (NB / 4); i += 256) {
            int r = i >> 5, c4 = (i & 31) << 2;
            const float* g = x + (long)(d0 + r) * N + n0 + c4;
            __builtin_amdgcn_global_load_async_to_lds_b128(
                (gv4p)(void*)g, (lv4p)&lX[r * LNB + c4], 0, 0);
        }
# if __has_builtin(__builtin_amdgcn_s_wait_asynccnt)
        __builtin_amdgcn_s_wait_asynccnt(0);
# else
        asm volatile("s_wait_asynccnt 0x0" ::: "memory");
# endif
#else
        // ---- fallback fill: coalesced b128 global loads -> ds_store_b128 ----
        for (int i = tid; i < KROWS * (NB / 4); i += 256) {
            int r = i >> 5, c4 = (i & 31) << 2;
            const float* g = a_bar + (long)r * N + n0 + c4;
            float4 v = *(const float4*)g;
            if (blk + 1 < blocksPerChunk) __builtin_prefetch(g + NB, 0, 1);
            *(float4*)&lA[r * LNB + c4] = v;
        }
        for (int i = tid; i < TILE_D * (NB / 4); i += 256) {
            int r = i >> 5, c4 = (i & 31) << 2;
            const float* g = x + (long)(d0 + r) * N + n0 + c4;
            float4 v = *(const float4*)g;
            if (blk + 1 < blocksPerChunk) __builtin_prefetch(g + NB, 0, 1);
            *(float4*)&lX[r * LNB + c4] = v;
        }
#endif
        __syncthreads();

        // ---- WMMA over NB/4 = 32 n-steps ----
        // A frag: lane L, vgpr v <-> (m = L%16, k = 2*(L/16)+v); B symmetric.
        const float* aFrag = &lA[(kt * 16 + llo) * LNB + lhi * 2];
        const float* bFrag = &lX[llo * LNB + lhi * 2];
        #pragma unroll 4
        for (int s = 0; s < NB / 4; ++s) {
            v2f af = *(const v2f*)(aFrag + s * 4);
            v2f b0 = *(const v2f*)(bFrag + s * 4 + (dquad + 0) * 16 * LNB);
            v2f b1 = *(const v2f*)(bFrag + s * 4 + (dquad + 1) * 16 * LNB);
            v2f b2 = *(const v2f*)(bFrag + s * 4 + (dquad + 2) * 16 * LNB);
            v2f b3 = *(const v2f*)(bFrag + s * 4 + (dquad + 3) * 16 * LNB);
            acc0 = __builtin_amdgcn_wmma_f32_16x16x4_f32(false, af, false, b0, (short)0, acc0, false, false);
            acc1 = __builtin_amdgcn_wmma_f32_16x16x4_f32(false, af, false, b1, (short)0, acc1, false, false);
            acc2 = __builtin_amdgcn_wmma_f32_16x16x4_f32(false, af, false, b2, (short)0, acc2, false, false);
            acc3 = __builtin_amdgcn_wmma_f32_16x16x4_f32(false, af, false, b3, (short)0, acc3, false, false);
        }

        // ---- a_bar row partial sums (only one D-tile per chunk does this) ----
        if (dtile == 0) {
            const float* rs = &lA[(tid & 63) * LNB + (tid >> 6) * 32];
            float s = 0.0f;
            #pragma unroll
            for (int j = 0; j < 32; ++j) s += rs[j];
            rowsum += s;
        }
        __syncthreads();
    }

    // ---- write partial tile: D-layout lane L, vgpr v -> (k = kt*16+v+8*lhi, d = llo) ----
    {
        float* pbase = partial + (long)chunk * (KROWS * DCOLS)
                     + (long)(kt * 16 + 8 * lhi) * DCOLS + d0 + llo;
        #pragma unroll
        for (int v = 0; v < 8; ++v) pbase[(long)v * DCOLS + (dquad + 0) * 16] = acc0[v];
        #pragma unroll
        for (int v = 0; v < 8; ++v) pbase[(long)v * DCOLS + (dquad + 1) * 16] = acc1[v];
        #pragma unroll
        for (int v = 0; v < 8; ++v) pbase[(long)v * DCOLS + (dquad + 2) * 16] = acc2[v];
        #pragma unroll
        for (int v = 0; v < 8; ++v) pbase[(long)v * DCOLS + (dquad + 3) * 16] = acc3[v];
    }

    if (dtile == 0) {
        lSum[tid] = rowsum;
        __syncthreads();
        if (tid < 64) {
            float s = lSum[tid] + lSum[tid + 64] + lSum[tid + 128] + lSum[tid + 192];
            asum_p[chunk * KROWS + tid] = s;
        }
    }
}

// ---------------------------------------------------------------------------
// Pass 2: out[k,d] = sum_s partial[s][k][d] - c[k,d] * sum_s asum_p[s][k]
// ---------------------------------------------------------------------------
__global__ __launch_bounds__(256) void vlad_finish(
    const float* __restrict__ partial, const float* __restrict__ asum_p,
    const float* __restrict__ c, float* __restrict__ out, int NS)
{
    int idx = blockIdx.x * 256 + threadIdx.x;
    if (idx >= KROWS * DCOLS) return;
    int k = idx >> 9;   // / DCOLS
    float acc = 0.0f;
    for (int s = 0; s < NS; ++s) acc += partial[(long)s * (KROWS * DCOLS) + idx];
    float as = 0.0f;
    for (int s = 0; s < NS; ++s) as += asum_p[s * KROWS + k];
    out[idx] = acc - c[idx] * as;
}

extern "C" void kernel_launch(void* const* d_in, const int* in_sizes, int n_in,
                              void* d_out, int out_size, void* d_ws, size_t ws_size,
                              hipStream_t stream) {
    const float* x     = (const float*)d_in[0];   // (512, N)
    const float* a_bar = (const float*)d_in[1];   // (64, N)
    const float* c     = (const float*)d_in[2];   // (64, 512)
    float* out = (float*)d_out;

    const int N = in_sizes[0] / DCOLS;            // 131072

    // Choose N-split so deterministic partials fit in workspace.
    int NS = 64;
    while (NS > 1 && (size_t)NS * (size_t)(KROWS * DCOLS + KROWS) * 4 > ws_size) NS >>= 1;
    while (NS > 1 && (N % (NS * NB)) != 0) NS >>= 1;
    const int blocksPerChunk = N / (NS * NB);

    float* partial = (float*)d_ws;
    float* asum_p  = partial + (size_t)NS * (KROWS * DCOLS);

    dim3 grid1(DCOLS / TILE_D, NS);
    vlad_gemm_wmma<<<grid1, 256, 0, stream>>>(x, a_bar, partial, asum_p, N, blocksPerChunk);

    int nOut = KROWS * DCOLS;
    vlad_finish<<<(nOut + 255) / 256, 256, 0, stream>>>(partial, asum_p, c, out, NS);
}